// GravNetBlock_3556232921273
// MI455X (gfx1250) — compile-verified
//
#include <hip/hip_runtime.h>
#include <math.h>

#define N_PTS 16384
#define C_DIM 128
#define S_DIM 4
#define P_DIM 32
#define K_NN  16
#define NPART 4
#define PART_SZ (N_PTS / NPART)   // 4096 candidates per partition

typedef float v2f __attribute__((ext_vector_type(2)));
typedef float v8f __attribute__((ext_vector_type(8)));

__device__ __forceinline__ v8f v8zero() {
  v8f z;
#pragma unroll
  for (int i = 0; i < 8; ++i) z[i] = 0.f;
  return z;
}

// D = A(16x4 f32) * B(4x16 f32) + C(16x16 f32)   -> v_wmma_f32_16x16x4_f32
__device__ __forceinline__ v8f wmma16x16x4(v2f a, v2f b, v8f c) {
  return __builtin_amdgcn_wmma_f32_16x16x4_f32(false, a, false, b, (short)0, c,
                                               false, false);
}

// ---------------------------------------------------------------------------
// Kernel 1: s = x@Ws + bs, h = x@Wh + bh  via one WMMA pass over [Ws|Wh]
// (128x48 concat weight tile in LDS, 3 column tiles of 16; cols 36..47 zero)
// block = 256 threads = 8 waves, each wave computes one 16-row tile.
// ---------------------------------------------------------------------------
__global__ void __launch_bounds__(256)
embed_kernel(const float* __restrict__ x, const float* __restrict__ Ws,
             const float* __restrict__ bs, const float* __restrict__ Wh,
             const float* __restrict__ bh, float* __restrict__ s_out,
             float* __restrict__ h_out) {
  __shared__ float Wc[128 * 48];  // 24 KB
  const int tid = threadIdx.x;
  for (int t = tid; t < 128 * 48; t += 256) {
    int c = t / 48, j = t - c * 48;
    float v = 0.f;
    if (j < 4)       v = Ws[c * 4 + j];
    else if (j < 36) v = Wh[c * 32 + (j - 4)];
    Wc[t] = v;
  }
  __syncthreads();

  const int wave = tid >> 5, lane = tid & 31;
  const int l15 = lane & 15, lhalf = lane >> 4, kb = lhalf * 2;
  const int r0 = (blockIdx.x * 8 + wave) * 16;

  v8f acc[3];
#pragma unroll
  for (int ct = 0; ct < 3; ++ct) acc[ct] = v8zero();

  const float* xrow = x + (size_t)(r0 + l15) * C_DIM + kb;
#pragma unroll 4
  for (int kk = 0; kk < 32; ++kk) {      // K = 128 in steps of 4
    v2f a;
    a.x = xrow[4 * kk];
    a.y = xrow[4 * kk + 1];
    const int kr = 4 * kk + kb;
#pragma unroll
    for (int ct = 0; ct < 3; ++ct) {
      v2f b;
      b.x = Wc[kr * 48 + ct * 16 + l15];
      b.y = Wc[(kr + 1) * 48 + ct * 16 + l15];
      acc[ct] = wmma16x16x4(a, b, acc[ct]);
    }
  }
#pragma unroll
  for (int ct = 0; ct < 3; ++ct) {
    const int col = ct * 16 + l15;
#pragma unroll
    for (int v = 0; v < 8; ++v) {
      const int row = r0 + v + 8 * lhalf;
      const float val = acc[ct][v];
      if (col < 4)
        s_out[row * 4 + col] = val + bs[col];
      else if (col < 36)
        h_out[row * 32 + (col - 4)] = val + bh[col - 4];
    }
  }
}

// ---------------------------------------------------------------------------
// Kernel 2a: partitioned brute-force kNN. Grid = (N/256 query chunks) x NPART.
// Each block scans one quarter of the candidate axis for 256 queries,
// keeping a register insertion-sorted top-16 of exact d2 per thread.
// 4x more waves in flight than a monolithic scan -> hides the LDS-load /
// fma / cmp dependency chain. Partial sorted lists go to the workspace.
// ---------------------------------------------------------------------------
__global__ void __launch_bounds__(256)
knn_part_kernel(const float* __restrict__ s, float* __restrict__ pd2,
                int* __restrict__ pidx) {
  __shared__ float4 tile[512];  // 8 KB
  const int qchunk = blockIdx.x >> 2;
  const int part = blockIdx.x & 3;
  const int i = qchunk * 256 + threadIdx.x;
  const int c0 = part * PART_SZ;
  const float4 si = ((const float4*)s)[i];

  float bd[K_NN];
  int bi[K_NN];
#pragma unroll
  for (int k = 0; k < K_NN; ++k) { bd[k] = 3.4e38f; bi[k] = 0; }

  for (int t0 = 0; t0 < PART_SZ; t0 += 512) {
    __syncthreads();
    tile[threadIdx.x]       = ((const float4*)s)[c0 + t0 + threadIdx.x];
    tile[threadIdx.x + 256] = ((const float4*)s)[c0 + t0 + threadIdx.x + 256];
    if (t0 + 512 < PART_SZ)  // prefetch next tile -> global_prefetch_b8
      __builtin_prefetch((const void*)&((const float4*)s)[c0 + t0 + 512 +
                                                          threadIdx.x], 0, 3);
    __syncthreads();
#pragma unroll 4
    for (int j = 0; j < 512; ++j) {   // LDS broadcast across lanes
      const float4 sj = tile[j];
      const float dx = si.x - sj.x, dy = si.y - sj.y;
      const float dz = si.z - sj.z, dw = si.w - sj.w;
      const float d2 = dx * dx + dy * dy + dz * dz + dw * dw;
      if (d2 < bd[K_NN - 1]) {       // rare after warmup
        bd[K_NN - 1] = d2;
        bi[K_NN - 1] = c0 + t0 + j;
#pragma unroll
        for (int m = K_NN - 1; m > 0; --m) {   // single bubble pass
          if (bd[m] < bd[m - 1]) {             // strict < : stable ties
            float tf = bd[m - 1]; bd[m - 1] = bd[m]; bd[m] = tf;
            int   ti = bi[m - 1]; bi[m - 1] = bi[m]; bi[m] = ti;
          }
        }
      }
    }
  }
  // layout [i][part][k]: merge kernel reads 64 contiguous entries per query
  float* pd = pd2 + ((size_t)i * NPART + part) * K_NN;
  int*   pi = pidx + ((size_t)i * NPART + part) * K_NN;
#pragma unroll
  for (int k = 0; k < K_NN; ++k) { pd[k] = bd[k]; pi[k] = bi[k]; }
}

// ---------------------------------------------------------------------------
// Kernel 2b: merge NPART sorted partial top-16 lists into the final top-16
// (register insertion over 64 entries; final set is order-invariant for the
// mean/max aggregation). Also emits edge weights w = exp(-10*d2).
// ---------------------------------------------------------------------------
__global__ void __launch_bounds__(256)
knn_merge_kernel(const float* __restrict__ pd2, const int* __restrict__ pidx,
                 int* __restrict__ idx_out, float* __restrict__ w_out) {
  const int i = blockIdx.x * 256 + threadIdx.x;
  const float* pd = pd2 + (size_t)i * NPART * K_NN;
  const int*   pi = pidx + (size_t)i * NPART * K_NN;

  float bd[K_NN];
  int bi[K_NN];
#pragma unroll
  for (int k = 0; k < K_NN; ++k) { bd[k] = 3.4e38f; bi[k] = 0; }

  for (int e = 0; e < NPART * K_NN; ++e) {
    const float d2 = pd[e];
    const int   j  = pi[e];
    if (d2 < bd[K_NN - 1]) {
      bd[K_NN - 1] = d2;
      bi[K_NN - 1] = j;
#pragma unroll
      for (int m = K_NN - 1; m > 0; --m) {
        if (bd[m] < bd[m - 1]) {
          float tf = bd[m - 1]; bd[m - 1] = bd[m]; bd[m] = tf;
          int   ti = bi[m - 1]; bi[m - 1] = bi[m]; bi[m] = ti;
        }
      }
    }
  }
#pragma unroll
  for (int k = 0; k < K_NN; ++k) {
    idx_out[i * K_NN + k] = bi[k];
    w_out[i * K_NN + k]   = expf(-10.f * bd[k]);
  }
}

// ---------------------------------------------------------------------------
// Kernel 3: weighted mean/max aggregation. One wave per query, lane = feature.
// agg[i] = [mean(h[idx]*w) (32) | max(h[idx]*w) (32)]
// ---------------------------------------------------------------------------
__global__ void __launch_bounds__(256)
agg_kernel(const float* __restrict__ h, const int* __restrict__ idx,
           const float* __restrict__ w, float* __restrict__ agg) {
  const int wave = threadIdx.x >> 5, lane = threadIdx.x & 31;
  const int i = blockIdx.x * 8 + wave;
  float sum = 0.f, mx = -3.4e38f;
#pragma unroll
  for (int k = 0; k < K_NN; ++k) {
    const int j = idx[i * K_NN + k];       // same addr across lanes: broadcast
    const float wk = w[i * K_NN + k];
    const float v = h[(size_t)j * P_DIM + lane] * wk;  // coalesced
    sum += v;
    mx = fmaxf(mx, v);
  }
  agg[i * 64 + lane]      = sum * (1.f / 16.f);
  agg[i * 64 + 32 + lane] = mx;
}

// ---------------------------------------------------------------------------
// Kernel 4: fused  gn = x@Wo1 + agg@Wo2 + bo2 ; y = gn@Wl + bl ;
//           LayerNorm ; ReLU ; + x.   One wave per 16-row tile, all WMMA f32.
// gn tile staged in padded LDS for C-layout -> A-layout conversion.
// Weights (160 KB) stay L2-resident (192 MB L2).
// ---------------------------------------------------------------------------
__global__ void __launch_bounds__(128)
fused_out_kernel(const float* __restrict__ x, const float* __restrict__ agg,
                 const float* __restrict__ Wo1, const float* __restrict__ Wo2,
                 const float* __restrict__ bo2, const float* __restrict__ Wl,
                 const float* __restrict__ bl, const float* __restrict__ gamma,
                 const float* __restrict__ beta, float* __restrict__ out) {
  __shared__ float gn_s[4][16][132];  // 33 KB, pad 132 -> conflict-free
  const int tid = threadIdx.x;
  const int wave = tid >> 5, lane = tid & 31;
  const int l15 = lane & 15, lhalf = lane >> 4, kb = lhalf * 2;
  const int r0 = (blockIdx.x * 4 + wave) * 16;

  v8f acc[8];
#pragma unroll
  for (int ct = 0; ct < 8; ++ct) acc[ct] = v8zero();

  // gn += x @ Wo1   (K = 128)
  const float* xrow = x + (size_t)(r0 + l15) * C_DIM + kb;
#pragma unroll 2
  for (int kk = 0; kk < 32; ++kk) {
    v2f a;
    a.x = xrow[4 * kk];
    a.y = xrow[4 * kk + 1];
    const int kr = 4 * kk + kb;
#pragma unroll
    for (int ct = 0; ct < 8; ++ct) {
      v2f b;
      b.x = Wo1[kr * 128 + ct * 16 + l15];
      b.y = Wo1[(kr + 1) * 128 + ct * 16 + l15];
      acc[ct] = wmma16x16x4(a, b, acc[ct]);
    }
  }
  // gn += agg @ Wo2   (K = 64)
  const float* arow = agg + (size_t)(r0 + l15) * 64 + kb;
#pragma unroll 2
  for (int kk = 0; kk < 16; ++kk) {
    v2f a;
    a.x = arow[4 * kk];
    a.y = arow[4 * kk + 1];
    const int kr = 4 * kk + kb;
#pragma unroll
    for (int ct = 0; ct < 8; ++ct) {
      v2f b;
      b.x = Wo2[kr * 128 + ct * 16 + l15];
      b.y = Wo2[(kr + 1) * 128 + ct * 16 + l15];
      acc[ct] = wmma16x16x4(a, b, acc[ct]);
    }
  }
  // stage gn (+bo2) to LDS (per-wave buffer; DS ops are wave-wide & in-order)
#pragma unroll
  for (int ct = 0; ct < 8; ++ct) {
    const int col = ct * 16 + l15;
    const float bias = bo2[col];
#pragma unroll
    for (int v = 0; v < 8; ++v)
      gn_s[wave][v + 8 * lhalf][col] = acc[ct][v] + bias;
  }

  // y = gn @ Wl  (K = 128)
#pragma unroll
  for (int ct = 0; ct < 8; ++ct) acc[ct] = v8zero();
#pragma unroll 2
  for (int kk = 0; kk < 32; ++kk) {
    v2f a;
    a.x = gn_s[wave][l15][4 * kk + kb];
    a.y = gn_s[wave][l15][4 * kk + kb + 1];
    const int kr = 4 * kk + kb;
#pragma unroll
    for (int ct = 0; ct < 8; ++ct) {
      v2f b;
      b.x = Wl[kr * 128 + ct * 16 + l15];
      b.y = Wl[(kr + 1) * 128 + ct * 16 + l15];
      acc[ct] = wmma16x16x4(a, b, acc[ct]);
    }
  }

  // + bl, LayerNorm stats per row (rows v+8*lhalf live in 16-lane halves)
  float rsum[8], rsq[8];
#pragma unroll
  for (int v = 0; v < 8; ++v) { rsum[v] = 0.f; rsq[v] = 0.f; }
#pragma unroll
  for (int ct = 0; ct < 8; ++ct) {
    const float bias = bl[ct * 16 + l15];
#pragma unroll
    for (int v = 0; v < 8; ++v) {
      const float val = acc[ct][v] + bias;
      acc[ct][v] = val;
      rsum[v] += val;
      rsq[v] += val * val;
    }
  }
#pragma unroll
  for (int m = 1; m <= 8; m <<= 1) {  // butterfly within each 16-lane half
#pragma unroll
    for (int v = 0; v < 8; ++v) {
      rsum[v] += __shfl_xor(rsum[v], m, 32);
      rsq[v]  += __shfl_xor(rsq[v], m, 32);
    }
  }
  // normalize, ReLU, residual, store
#pragma unroll
  for (int ct = 0; ct < 8; ++ct) {
    const int col = ct * 16 + l15;
    const float g = gamma[col], bt = beta[col];
#pragma unroll
    for (int v = 0; v < 8; ++v) {
      const int row = r0 + v + 8 * lhalf;
      const float mu  = rsum[v] * (1.f / 128.f);
      const float var = rsq[v] * (1.f / 128.f) - mu * mu;
      const float yn = (acc[ct][v] - mu) * rsqrtf(var + 1e-5f) * g + bt;
      out[(size_t)row * C_DIM + col] =
          fmaxf(yn, 0.f) + x[(size_t)row * C_DIM + col];
    }
  }
}

// ---------------------------------------------------------------------------
extern "C" void kernel_launch(void* const* d_in, const int* in_sizes, int n_in,
                              void* d_out, int out_size, void* d_ws,
                              size_t ws_size, hipStream_t stream) {
  const float* x    = (const float*)d_in[0];
  const float* Ws   = (const float*)d_in[1];
  const float* bs   = (const float*)d_in[2];
  const float* Wh   = (const float*)d_in[3];
  const float* bh   = (const float*)d_in[4];
  const float* Wo1  = (const float*)d_in[5];
  const float* Wo2  = (const float*)d_in[6];
  const float* bo2  = (const float*)d_in[7];
  const float* Wl   = (const float*)d_in[8];
  const float* bl   = (const float*)d_in[9];
  const float* gm   = (const float*)d_in[10];
  const float* bt   = (const float*)d_in[11];

  float* ws  = (float*)d_ws;
  float* s   = ws;                                // N*4
  float* h   = ws + (size_t)N_PTS * 4;            // N*32
  int*   idx = (int*)(ws + (size_t)N_PTS * 36);   // N*16 ints
  float* w   = ws + (size_t)N_PTS * 52;           // N*16
  float* agg = ws + (size_t)N_PTS * 68;           // N*64
  float* pd2 = ws + (size_t)N_PTS * 132;          // N*4*16 partial d2
  int*  pidx = (int*)(ws + (size_t)N_PTS * 196);  // N*4*16 partial idx (~17MB)

  embed_kernel<<<N_PTS / (16 * 8), 256, 0, stream>>>(x, Ws, bs, Wh, bh, s, h);
  knn_part_kernel<<<(N_PTS / 256) * NPART, 256, 0, stream>>>(s, pd2, pidx);
  knn_merge_kernel<<<N_PTS / 256, 256, 0, stream>>>(pd2, pidx, idx, w);
  agg_kernel<<<N_PTS / 8, 256, 0, stream>>>(h, idx, w, agg);
  fused_out_kernel<<<N_PTS / (16 * 4), 128, 0, stream>>>(
      x, agg, Wo1, Wo2, bo2, Wl, bl, gm, bt, (float*)d_out);
}